// AttnDecoderRNN_56126632624693
// MI455X (gfx1250) — compile-verified
//
#include <hip/hip_runtime.h>
#include <math.h>

typedef __attribute__((ext_vector_type(2))) float v2f;
typedef __attribute__((ext_vector_type(4))) float v4f;
typedef __attribute__((ext_vector_type(8))) float v8f;

#define Hh   512
#define Vv   100000
#define Ll   15
#define Bb   64

// ---- d_out layout (floats) ----
#define OUT_LOGITS 0
#define OUT_H      6400000
#define OUT_C      6432768
#define OUT_AW     6465536

// ---- workspace layout (floats) ----
#define WS_UAB  0                     // 1024*512
#define WS_WAB  524288                // 512*512
#define WS_WPB  786432                // 512*512
#define WS_U    1048576               // 960*512
#define WS_HWA  1540096               // 64*512
#define WS_HWP  1572864               // 64*512
#define WS_HB0  1605632               // 64*512 arranged
#define WS_X    1638400               // 64*1536
#define WS_XB   1736704               // 64*1536 arranged
#define WS_G1   1835008               // 2048*64
#define WS_G2   1966080               // 2048*64
#define WS_HBN  2097152               // 64*512 arranged
#define WS_RS   2129920               // 64

static __device__ __forceinline__ v8f wmma4(v2f a, v2f b, v8f c) {
  return __builtin_amdgcn_wmma_f32_16x16x4_f32(false, a, false, b, (short)0, c,
                                               false, false);
}
static __device__ __forceinline__ float sigf(float x) {
  return 1.0f / (1.0f + __expf(-x));
}

// Arrange B from a [K,N] row-major source into WMMA-B register layout:
// Bws[((ct*KS+ks)*32+lane)*2+j] = src[(4*ks + 2*(lane>>4) + j)*N + ct*16 + (lane&15)]
__global__ void arr_kn_kernel(const float* __restrict__ src, float* __restrict__ dst,
                              int K, int N) {
  int tid  = blockIdx.x * blockDim.x + threadIdx.x;
  int lane = tid & 31;
  int r    = tid >> 5;
  int KS = K >> 2, NT = N >> 4;
  if (r >= KS * NT) return;
  int ks = r % KS, ct = r / KS;
  int k = 4 * ks + 2 * (lane >> 4);
  int n = ct * 16 + (lane & 15);
  v2f v;
  v.x = src[(long)k * N + n];
  v.y = src[(long)(k + 1) * N + n];
  *(v2f*)(dst + ((long)(ct * KS + ks) * 32 + lane) * 2) = v;
}

// Arrange B from a [64,K] row-major source, transposed (B[k][n] = X[n][k]).
__global__ void arr_t_kernel(const float* __restrict__ X, float* __restrict__ dst,
                             int K) {
  int tid  = blockIdx.x * blockDim.x + threadIdx.x;
  int lane = tid & 31;
  int r    = tid >> 5;
  int KS = K >> 2;
  if (r >= KS * 4) return;
  int ks = r % KS, ct = r / KS;
  v2f v = *(const v2f*)(X + (long)(ct * 16 + (lane & 15)) * K + 4 * ks + 2 * (lane >> 4));
  *(v2f*)(dst + ((long)(ct * KS + ks) * 32 + lane) * 2) = v;
}

// Generic f32 WMMA GEMM: one 16x16 C tile per wave. A row-major [M,K] (lda),
// B pre-arranged, C row-major [M,N] (ldc).
__global__ void gemm_wmma_kernel(const float* __restrict__ A, int lda,
                                 const float* __restrict__ Bws,
                                 float* __restrict__ C, int ldc,
                                 int Mtiles, int Ntiles, int K) {
  int tid  = blockIdx.x * blockDim.x + threadIdx.x;
  int lane = threadIdx.x & 31;
  int wid  = tid >> 5;
  if (wid >= Mtiles * Ntiles) return;
  int rt = wid / Ntiles, ct = wid % Ntiles;
  int m0 = rt * 16, n0 = ct * 16;
  int KS = K >> 2;
  const float* aptr = A + (long)(m0 + (lane & 15)) * lda + 2 * (lane >> 4);
  const float* bptr = Bws + ((long)ct * KS * 32 + lane) * 2;
  v8f acc = {};
  for (int ks = 0; ks < KS; ++ks) {
    v2f a = *(const v2f*)aptr;  aptr += 4;
    v2f b = *(const v2f*)bptr;  bptr += 64;
    acc = wmma4(a, b, acc);
  }
  float accs[8];
  *(v8f*)accs = acc;
  int row = m0 + 8 * (lane >> 4);
  int col = n0 + (lane & 15);
#pragma unroll
  for (int i = 0; i < 8; ++i) C[(long)(row + i) * ldc + col] = accs[i];
}

// Logits: C'[v,b] = out_W @ h_new^T. 5 v-tiles x 4 b-tiles per wave
// (1250 jobs x 80 vocab rows = 100000 exactly; B reuse x5 -> ~160MB L2 traffic).
// out_W streamed with non-temporal loads (read-once, 205MB).
// Epilogue: in-wave LDS transpose -> out[b][v] with out_b fused, b128 stores.
__global__ void logits_wmma_kernel(const float* __restrict__ Wout,
                                   const float* __restrict__ hB,
                                   const float* __restrict__ bias,
                                   float* __restrict__ out) {
  __shared__ float lds[8][16 * 17];
  int tid  = threadIdx.x;
  int lane = tid & 31;
  int w    = tid >> 5;
  int job  = blockIdx.x * 8 + w;
  if (job >= 1250) return;
  int v0 = job * 80;
  int lo = lane & 15, hi = lane >> 4;

  const float* ap[5];
#pragma unroll
  for (int vt = 0; vt < 5; ++vt)
    ap[vt] = Wout + (long)(v0 + vt * 16 + lo) * Hh + 2 * hi;
  const float* bb = hB + lane * 2;

  v8f acc[5][4];
  v8f z = {};
#pragma unroll
  for (int i = 0; i < 5; ++i)
#pragma unroll
    for (int j = 0; j < 4; ++j) acc[i][j] = z;

  for (int ks = 0; ks < 128; ++ks) {
    v2f A[5];
#pragma unroll
    for (int vt = 0; vt < 5; ++vt)
      A[vt] = __builtin_nontemporal_load((const v2f*)(ap[vt] + 4 * ks));
#pragma unroll
    for (int ct = 0; ct < 4; ++ct) {
      v2f B = *(const v2f*)(bb + (ct * 128 + ks) * 64);
#pragma unroll
      for (int vt = 0; vt < 5; ++vt)
        acc[vt][ct] = wmma4(A[vt], B, acc[vt][ct]);
    }
  }

  float* myl = &lds[w][0];
  int bl = lane >> 1;          // batch within tile for the store phase
  int voff = 8 * (lane & 1);   // v offset within tile
#pragma unroll
  for (int t = 0; t < 20; ++t) {
    int vt = t >> 2, ct = t & 3;
    float accs[8];
    *(v8f*)accs = acc[vt][ct];
#pragma unroll
    for (int i = 0; i < 8; ++i) myl[(i + 8 * hi) * 17 + lo] = accs[i];
    __asm__ volatile("s_wait_dscnt 0x0" ::: "memory");
    int vbase = v0 + vt * 16 + voff;
    float o[8];
#pragma unroll
    for (int i = 0; i < 8; ++i)
      o[i] = myl[(voff + i) * 17 + bl] + bias[vbase + i];
    long addr = (long)(ct * 16 + bl) * Vv + vbase;
    *(v4f*)(out + OUT_LOGITS + addr)     = *(v4f*)&o[0];
    *(v4f*)(out + OUT_LOGITS + addr + 4) = *(v4f*)&o[4];
    __asm__ volatile("s_wait_dscnt 0x0" ::: "memory");
  }
}

// Attention: scores, p_t, gaussian-softmax, attn_out; builds x = [embed|attn_out].
__global__ void attn_kernel(const int* __restrict__ ids,
                            const float* __restrict__ enc,
                            const float* __restrict__ emb,
                            const float* __restrict__ U,
                            const float* __restrict__ hWa,
                            const float* __restrict__ hWp,
                            const float* __restrict__ ba,
                            const float* __restrict__ Vp,
                            float* __restrict__ x,
                            float* __restrict__ out) {
  int b = blockIdx.x;
  int tid = threadIdx.x, w = tid >> 5, lane = tid & 31;
  __shared__ float s_sc[16];
  __shared__ float s_pt;
  __shared__ float s_aw[16];

  if (w < Ll) {
    const float* hw = hWa + b * Hh;
    const float* u  = U + (long)(w * Bb + b) * Hh;
    float p = 0.0f;
    for (int n = lane; n < Hh; n += 32) p += ba[n] * tanhf(hw[n] + u[n]);
    for (int off = 16; off; off >>= 1) p += __shfl_down(p, off, 32);
    if (lane == 0) s_sc[w] = p;
  } else if (w == 15) {
    const float* hp = hWp + b * Hh;
    float p = 0.0f;
    for (int n = lane; n < Hh; n += 32) p += Vp[n] * tanhf(hp[n]);
    for (int off = 16; off; off >>= 1) p += __shfl_down(p, off, 32);
    if (lane == 0) s_pt = 15.0f * sigf(p);
  }
  __syncthreads();
  if (w == 0) {
    float pt = s_pt;
    float sc = -3.0e38f;
    if (lane < Ll) {
      float d = (float)lane - pt;
      sc = s_sc[lane] * __expf(d * d * (1.0f / 6.25f));  // (D/2)^2 = 6.25
    }
    float m = sc;
    for (int off = 16; off; off >>= 1) m = fmaxf(m, __shfl_xor(m, off, 32));
    float e = (lane < Ll) ? __expf(sc - m) : 0.0f;
    float s = e;
    for (int off = 16; off; off >>= 1) s += __shfl_xor(s, off, 32);
    if (lane < Ll) {
      float aw = e / s;
      s_aw[lane] = aw;
      out[OUT_AW + b * Ll + lane] = aw;
    }
  }
  __syncthreads();
  // x[:, 0:512] = embedding row
  {
    long id = (long)ids[b];
    if (tid < Hh) x[(long)b * 1536 + tid] = emb[id * Hh + tid];
  }
  // x[:, 512:1536] = attn_out (faithful raw-reshape indexing)
  for (int j = tid; j < 1024; j += blockDim.x) {
    float a = 0.0f;
#pragma unroll
    for (int l = 0; l < Ll; ++l)
      a += s_aw[l] * enc[(long)b * 15360 + l * 1024 + j];
    x[(long)b * 1536 + Hh + j] = a;
  }
}

__global__ void lstm_kernel(const float* __restrict__ g1, const float* __restrict__ g2,
                            const float* __restrict__ bih, const float* __restrict__ bhh,
                            const float* __restrict__ c_in, float* __restrict__ out) {
  int t = blockIdx.x * blockDim.x + threadIdx.x;
  if (t >= Bb * Hh) return;
  int b = t & 63, u = t >> 6;
  float ig = g1[u * 64 + b]            + g2[u * 64 + b]            + bih[u]        + bhh[u];
  float fg = g1[(u + 512) * 64 + b]    + g2[(u + 512) * 64 + b]    + bih[u + 512]  + bhh[u + 512];
  float gg = g1[(u + 1024) * 64 + b]   + g2[(u + 1024) * 64 + b]   + bih[u + 1024] + bhh[u + 1024];
  float og = g1[(u + 1536) * 64 + b]   + g2[(u + 1536) * 64 + b]   + bih[u + 1536] + bhh[u + 1536];
  float cp = c_in[b * Hh + u];
  float cn = sigf(fg) * cp + sigf(ig) * tanhf(gg);
  float hn = sigf(og) * tanhf(cn);
  out[OUT_H + b * Hh + u] = hn;
  out[OUT_C + b * Hh + u] = cn;
}

// Per-row online logsumexp over V.
__global__ void lse_kernel(const float* __restrict__ out, float* __restrict__ rs) {
  int b = blockIdx.x, tid = threadIdx.x;
  const float* row = out + OUT_LOGITS + (long)b * Vv;
  float m = -3.0e38f, s = 0.0f;
  for (int v = tid; v < Vv; v += 256) {
    float xv = row[v];
    if (xv > m) { s = s * __expf(m - xv) + 1.0f; m = xv; }
    else        { s += __expf(xv - m); }
  }
  for (int off = 16; off; off >>= 1) {
    float m2 = __shfl_down(m, off, 32);
    float s2 = __shfl_down(s, off, 32);
    float M = fmaxf(m, m2);
    s = s * __expf(m - M) + s2 * __expf(m2 - M);
    m = M;
  }
  __shared__ float sm[8], ss[8];
  if ((tid & 31) == 0) { sm[tid >> 5] = m; ss[tid >> 5] = s; }
  __syncthreads();
  if (tid == 0) {
    float M = sm[0], S = ss[0];
    for (int i = 1; i < 8; ++i) {
      float MM = fmaxf(M, sm[i]);
      S = S * __expf(M - MM) + ss[i] * __expf(sm[i] - MM);
      M = MM;
    }
    rs[b] = M + __logf(S);
  }
}

__global__ void sub_kernel(float* __restrict__ out, const float* __restrict__ rs) {
  int b = blockIdx.y;
  int v = blockIdx.x * blockDim.x + threadIdx.x;
  if (v < Vv) out[OUT_LOGITS + (long)b * Vv + v] -= rs[b];
}

extern "C" void kernel_launch(void* const* d_in, const int* in_sizes, int n_in,
                              void* d_out, int out_size, void* d_ws, size_t ws_size,
                              hipStream_t stream) {
  const int*   ids  = (const int*)d_in[0];
  const float* hid  = (const float*)d_in[1];
  const float* cin  = (const float*)d_in[2];
  const float* enc  = (const float*)d_in[3];
  const float* emb  = (const float*)d_in[4];
  const float* Wa   = (const float*)d_in[5];
  const float* Ua   = (const float*)d_in[6];
  const float* ba   = (const float*)d_in[7];
  const float* Wp   = (const float*)d_in[8];
  const float* Vp   = (const float*)d_in[9];
  const float* Wih  = (const float*)d_in[10];
  const float* Whh  = (const float*)d_in[11];
  const float* bih  = (const float*)d_in[12];
  const float* bhh  = (const float*)d_in[13];
  const float* Wout = (const float*)d_in[14];
  const float* bout = (const float*)d_in[15];
  float* out = (float*)d_out;
  float* ws  = (float*)d_ws;

  // 1) Arrange static B operands into WMMA layout
  arr_kn_kernel<<<1024, 256, 0, stream>>>(Ua, ws + WS_UAB, 1024, 512);
  arr_kn_kernel<<<512, 256, 0, stream>>>(Wa, ws + WS_WAB, 512, 512);
  arr_kn_kernel<<<512, 256, 0, stream>>>(Wp, ws + WS_WPB, 512, 512);
  arr_t_kernel <<<64, 256, 0, stream>>>(hid, ws + WS_HB0, 512);

  // 2) U = enc @ Ua  (960x512),  hWa = h @ Wa,  hWp = h @ W_p
  gemm_wmma_kernel<<<240, 256, 0, stream>>>(enc, 1024, ws + WS_UAB, ws + WS_U, 512, 60, 32, 1024);
  gemm_wmma_kernel<<<16, 256, 0, stream>>>(hid, 512, ws + WS_WAB, ws + WS_HWA, 512, 4, 32, 512);
  gemm_wmma_kernel<<<16, 256, 0, stream>>>(hid, 512, ws + WS_WPB, ws + WS_HWP, 512, 4, 32, 512);

  // 3) Attention + x = [embed | attn_out]
  attn_kernel<<<64, 512, 0, stream>>>(ids, enc, emb, ws + WS_U, ws + WS_HWA,
                                      ws + WS_HWP, ba, Vp, ws + WS_X, out);

  // 4) Gates: g1 = W_ih @ x^T, g2 = W_hh @ h^T  (stored [2048 x 64])
  arr_t_kernel<<<192, 256, 0, stream>>>(ws + WS_X, ws + WS_XB, 1536);
  gemm_wmma_kernel<<<64, 256, 0, stream>>>(Wih, 1536, ws + WS_XB, ws + WS_G1, 64, 128, 4, 1536);
  gemm_wmma_kernel<<<64, 256, 0, stream>>>(Whh, 512, ws + WS_HB0, ws + WS_G2, 64, 128, 4, 512);

  // 5) LSTM cell -> h_new, c_new in d_out
  lstm_kernel<<<128, 256, 0, stream>>>(ws + WS_G1, ws + WS_G2, bih, bhh, cin, out);

  // 6) Logits = out_W @ h_new^T + out_b, transposed-stored as out[b][v]
  arr_t_kernel<<<64, 256, 0, stream>>>(out + OUT_H, ws + WS_HBN, 512);
  logits_wmma_kernel<<<157, 256, 0, stream>>>(Wout, ws + WS_HBN, bout, out);

  // 7) log_softmax: per-row logsumexp, then subtract
  lse_kernel<<<64, 256, 0, stream>>>(out, ws + WS_RS);
  sub_kernel<<<dim3(391, 64), 256, 0, stream>>>(out, ws + WS_RS);
}